// GraphEmbedder_34248069218772
// MI455X (gfx1250) — compile-verified
//
#include <hip/hip_runtime.h>
#include <hip/hip_bf16.h>

#define N_NODES   100000
#define N_EDGES   1000000
#define HID       64
#define EDGE_D    32
#define N_GRAPHS  256
#define NEG_SLOPE 0.2f

typedef __attribute__((ext_vector_type(2))) float v2f;
typedef __attribute__((ext_vector_type(8))) float v8f;

// ---------------------------------------------------------------------------
// float atomic max without CAS loop: sign-split int/uint ordering trick.
// Valid given init value -FLT_MAX.
// ---------------------------------------------------------------------------
__device__ __forceinline__ float atomicMaxF(float* addr, float val) {
  if (val >= 0.0f)
    return __int_as_float(atomicMax((int*)addr, __float_as_int(val)));
  else
    return __uint_as_float(atomicMin((unsigned int*)addr, __float_as_uint(val)));
}

// ---------------------------------------------------------------------------
// Prep: wv = We @ att_e (32), ws = W @ att_s (64), wd = W @ att_d (64)
// One block of 64 threads; tiny.
// ---------------------------------------------------------------------------
__global__ void gat_prep(const float* __restrict__ W,  const float* __restrict__ att_s,
                         const float* __restrict__ att_d,
                         const float* __restrict__ We, const float* __restrict__ att_e,
                         float* __restrict__ wsv, float* __restrict__ wdv,
                         float* __restrict__ wvv) {
  int k = threadIdx.x;          // 0..63
  float s = 0.f, d = 0.f;
  for (int c = 0; c < HID; ++c) {
    float w = W[k * HID + c];
    s += w * att_s[c];
    d += w * att_d[c];
  }
  wsv[k] = s;
  wdv[k] = d;
  if (k < EDGE_D) {
    float v = 0.f;
    for (int c = 0; c < HID; ++c) v += We[k * HID + c] * att_e[c];
    wvv[k] = v;
  }
}

// ---------------------------------------------------------------------------
// Per-layer init: acc = 0, mmax = -FLT_MAX, ssum = 0. Grid covers N*64.
// ---------------------------------------------------------------------------
__global__ void gat_init(float* __restrict__ acc, float* __restrict__ mmax,
                         float* __restrict__ ssum, int nN) {
  int i = blockIdx.x * blockDim.x + threadIdx.x;
  if (i < nN * HID) acc[i] = 0.0f;
  if (i < nN) { mmax[i] = -3.0e38f; ssum[i] = 0.0f; }
}

// ---------------------------------------------------------------------------
// WMMA node GEMM:  H[16-row tile] = X_tile(16x64) @ W(64x64), one wave/tile.
// Uses V_WMMA_F32_16X16X4_F32, 16 K-steps, 4 N-tiles -> 64 WMMA per wave.
// Fused: a_s = X @ wsv, a_d = X @ wdv (per-row, half-wave reduction).
// Wave32 layouts per CDNA5 ISA 7.12.2:
//   A 16x4:  lanes 0-15 row M=lane, v0/v1 = K0/K1; lanes 16-31 = K2/K3
//   B 4x16:  lanes 0-15 col N=lane, v0/v1 = K0/K1; lanes 16-31 = K2/K3
//   C 16x16: VGPR v: M=v (lanes 0-15), M=v+8 (lanes 16-31), N=lane%16
// ---------------------------------------------------------------------------
__global__ void gat_gemm_node(const float* __restrict__ X, const float* __restrict__ W,
                              const float* __restrict__ wsv, const float* __restrict__ wdv,
                              float* __restrict__ H, float* __restrict__ a_s,
                              float* __restrict__ a_d, int nRows) {
  __shared__ float Wl[HID * HID];
  __shared__ float wsl[HID];
  __shared__ float wdl[HID];
  for (int i = threadIdx.x; i < HID * HID; i += blockDim.x) Wl[i] = W[i];
  for (int i = threadIdx.x; i < HID; i += blockDim.x) { wsl[i] = wsv[i]; wdl[i] = wdv[i]; }
  __syncthreads();

  const int wave = threadIdx.x >> 5;
  const int lane = threadIdx.x & 31;
  const int half = lane >> 4;      // 0: K0/K1 half, 1: K2/K3 half
  const int l    = lane & 15;

  const int tile = blockIdx.x * (blockDim.x >> 5) + wave;
  const int row0 = tile * 16;
  if (row0 >= nRows) return;

  v8f acc[4];
  acc[0] = (v8f){}; acc[1] = (v8f){}; acc[2] = (v8f){}; acc[3] = (v8f){};
  float asacc = 0.f, adacc = 0.f;

  const float* xrow = X + (long)(row0 + l) * HID;

#pragma unroll
  for (int t = 0; t < 16; ++t) {
    const int ka = t * 4 + half * 2;
    v2f a = *(const v2f*)(xrow + ka);                   // A fragment (2 f32/lane)
    asacc += a.x * wsl[ka] + a.y * wsl[ka + 1];
    adacc += a.x * wdl[ka] + a.y * wdl[ka + 1];
#pragma unroll
    for (int n = 0; n < 4; ++n) {
      v2f b;
      b.x = Wl[ka * HID + n * 16 + l];
      b.y = Wl[(ka + 1) * HID + n * 16 + l];
      acc[n] = __builtin_amdgcn_wmma_f32_16x16x4_f32(
          false, a, false, b, (short)0, acc[n], false, false);
    }
  }

  // fused attention logits: combine the two K-halves of each row
  float astot = asacc + __shfl_xor(asacc, 16, 32);
  float adtot = adacc + __shfl_xor(adacc, 16, 32);
  if (half == 0) {
    a_s[row0 + l] = astot;
    a_d[row0 + l] = adtot;
  }

  // store 16x64 tile
#pragma unroll
  for (int n = 0; n < 4; ++n) {
#pragma unroll
    for (int v = 0; v < 8; ++v) {
      H[(long)(row0 + half * 8 + v) * HID + n * 16 + l] = acc[n][v];
    }
  }
}

// ---------------------------------------------------------------------------
// Edge pass 1: t = leaky(a_s[src] + a_d[dst] + edge_attr . wv); mmax via atomic.
// 8 lanes per edge (edge row = 32 f32 = 8 float4). E*8 threads, no tail wave.
// ---------------------------------------------------------------------------
__global__ void gat_edge_logits(const float* __restrict__ EA, const float* __restrict__ wv,
                                const int* __restrict__ src, const int* __restrict__ dst,
                                const float* __restrict__ a_s, const float* __restrict__ a_d,
                                float* __restrict__ tbuf, float* __restrict__ mmax, int nE) {
  const int gid = blockIdx.x * blockDim.x + threadIdx.x;
  const int li  = gid & 7;
  const int e   = gid >> 3;
  if (e >= nE) return;
  const float4 v = ((const float4*)(EA + (long)e * EDGE_D))[li];
  const float4 w = ((const float4*)wv)[li];
  float p = v.x * w.x + v.y * w.y + v.z * w.z + v.w * w.w;
  p += __shfl_xor(p, 1, 32);
  p += __shfl_xor(p, 2, 32);
  p += __shfl_xor(p, 4, 32);
  if (li == 0) {
    const int s = src[e], d = dst[e];
    float t = p + a_s[s] + a_d[d];
    t = (t > 0.f) ? t : t * NEG_SLOPE;
    tbuf[e] = t;
    atomicMaxF(&mmax[d], t);
  }
}

// ---------------------------------------------------------------------------
// Edge pass 2: ssum[dst] += exp(t - mmax[dst]). One thread per edge.
// ---------------------------------------------------------------------------
__global__ void gat_edge_expsum(const float* __restrict__ tbuf, const int* __restrict__ dst,
                                const float* __restrict__ mmax, float* __restrict__ ssum,
                                int nE) {
  const int e = blockIdx.x * blockDim.x + threadIdx.x;
  if (e >= nE) return;
  const int d = dst[e];
  atomicAdd(&ssum[d], __expf(tbuf[e] - mmax[d]));
}

// ---------------------------------------------------------------------------
// Edge pass 3: acc[dst] += h[src] * exp(t-m)/(s+eps). 16 lanes/edge, float4
// gathers + 4 scalar f32 atomics/lane (L2-resident targets). E*16 threads.
// ---------------------------------------------------------------------------
__global__ void gat_edge_aggregate(const float* __restrict__ tbuf, const int* __restrict__ src,
                                   const int* __restrict__ dst, const float* __restrict__ mmax,
                                   const float* __restrict__ ssum, const float* __restrict__ H,
                                   float* __restrict__ acc, int nE) {
  const int gid = blockIdx.x * blockDim.x + threadIdx.x;
  const int li  = gid & 15;
  const int e   = gid >> 4;
  if (e >= nE) return;
  const int s = src[e], d = dst[e];
  const float wgt = __expf(tbuf[e] - mmax[d]) / (ssum[d] + 1e-16f);
  const float4 h4 = ((const float4*)(H + (long)s * HID))[li];
  float* ap = acc + (long)d * HID + li * 4;
  atomicAdd(ap + 0, h4.x * wgt);
  atomicAdd(ap + 1, h4.y * wgt);
  atomicAdd(ap + 2, h4.z * wgt);
  atomicAdd(ap + 3, h4.w * wgt);
}

// ---------------------------------------------------------------------------
// hout = acc + bias (layer output / next-layer input). Grid covers N*64.
// ---------------------------------------------------------------------------
__global__ void gat_bias_copy(const float* __restrict__ acc, const float* __restrict__ b,
                              float* __restrict__ hout, int total) {
  const int i = blockIdx.x * blockDim.x + threadIdx.x;
  if (i >= total) return;
  hout[i] = acc[i] + b[i & (HID - 1)];
}

// ---------------------------------------------------------------------------
// Pooling
// ---------------------------------------------------------------------------
__global__ void pool_init(float* __restrict__ pooled, float* __restrict__ counts) {
  const int i = blockIdx.x * blockDim.x + threadIdx.x;
  if (i < N_GRAPHS * HID) pooled[i] = 0.0f;
  if (i < N_GRAPHS) counts[i] = 0.0f;
}

// final node feature = hprev + acc + b1 (residual), pooled per graph.
__global__ void gat_pool(const float* __restrict__ hprev, const float* __restrict__ acc,
                         const float* __restrict__ b1, const int* __restrict__ batch,
                         float* __restrict__ pooled, float* __restrict__ counts, int nN) {
  const int gid = blockIdx.x * blockDim.x + threadIdx.x;
  const int li  = gid & 15;
  const int n   = gid >> 4;
  if (n >= nN) return;
  const int g = batch[n];
  const float4 hp = ((const float4*)(hprev + (long)n * HID))[li];
  const float4 ac = ((const float4*)(acc   + (long)n * HID))[li];
  const float4 bb = ((const float4*)b1)[li];
  float* pp = pooled + (long)g * HID + li * 4;
  atomicAdd(pp + 0, hp.x + ac.x + bb.x);
  atomicAdd(pp + 1, hp.y + ac.y + bb.y);
  atomicAdd(pp + 2, hp.z + ac.z + bb.z);
  atomicAdd(pp + 3, hp.w + ac.w + bb.w);
  if (li == 0) atomicAdd(&counts[g], 1.0f);
}

__global__ void pool_final(const float* __restrict__ pooled, const float* __restrict__ counts,
                           float* __restrict__ out) {
  const int i = blockIdx.x * blockDim.x + threadIdx.x;
  if (i >= N_GRAPHS * HID) return;
  out[i] = pooled[i] / fmaxf(counts[i >> 6], 1.0f);
}

// ---------------------------------------------------------------------------
extern "C" void kernel_launch(void* const* d_in, const int* in_sizes, int n_in,
                              void* d_out, int out_size, void* d_ws, size_t ws_size,
                              hipStream_t stream) {
  const float* x      = (const float*)d_in[0];
  const int*   ei     = (const int*)d_in[1];      // [2, E] row-major
  const float* eattr  = (const float*)d_in[2];
  const int*   batch  = (const int*)d_in[3];
  const float* W0     = (const float*)d_in[4];
  const float* as0    = (const float*)d_in[5];
  const float* ad0    = (const float*)d_in[6];
  const float* We0    = (const float*)d_in[7];
  const float* ae0    = (const float*)d_in[8];
  const float* b0     = (const float*)d_in[9];
  const float* W1     = (const float*)d_in[10];
  const float* as1    = (const float*)d_in[11];
  const float* ad1    = (const float*)d_in[12];
  const float* We1    = (const float*)d_in[13];
  const float* ae1    = (const float*)d_in[14];
  const float* b1     = (const float*)d_in[15];
  float* out          = (float*)d_out;

  const int*   srcI = ei;
  const int*   dstI = ei + N_EDGES;

  const int NH = N_NODES * HID;
  float* w      = (float*)d_ws;
  float* hA     = w;                 // N*64  current-layer GEMM output
  float* acc    = hA   + NH;         // N*64  aggregation accumulator
  float* hprev  = acc  + NH;         // N*64  layer-0 output (residual input)
  float* a_s    = hprev + NH;        // N
  float* a_d    = a_s  + N_NODES;    // N
  float* mmax   = a_d  + N_NODES;    // N
  float* ssum   = mmax + N_NODES;    // N
  float* tbuf   = ssum + N_NODES;    // E
  float* wvv    = tbuf + N_EDGES;    // 32
  float* wsv    = wvv  + 64;         // 64
  float* wdv    = wsv  + 64;         // 64
  float* pooled = wdv  + 64;         // 256*64
  float* counts = pooled + N_GRAPHS * HID; // 256

  const int BT        = 256;
  const int gInit     = (NH + BT - 1) / BT;                       // 25000
  const int tiles     = N_NODES / 16;                             // 6250 (exact)
  const int gGemm     = (tiles + (BT / 32) - 1) / (BT / 32);      // 782
  const int gE8       = (N_EDGES * 8) / BT;                       // exact
  const int gE1       = (N_EDGES + BT - 1) / BT;
  const int gE16      = (N_EDGES * 16) / BT;                      // exact
  const int gN16      = (N_NODES * 16) / BT;                      // exact
  const int gPoolInit = (N_GRAPHS * HID + BT - 1) / BT;
  const int gFinal    = (N_GRAPHS * HID + BT - 1) / BT;

  // ---------------- Layer 0 ----------------
  gat_prep<<<1, 64, 0, stream>>>(W0, as0, ad0, We0, ae0, wsv, wdv, wvv);
  gat_init<<<gInit, BT, 0, stream>>>(acc, mmax, ssum, N_NODES);
  gat_gemm_node<<<gGemm, BT, 0, stream>>>(x, W0, wsv, wdv, hA, a_s, a_d, N_NODES);
  gat_edge_logits<<<gE8, BT, 0, stream>>>(eattr, wvv, srcI, dstI, a_s, a_d, tbuf, mmax, N_EDGES);
  gat_edge_expsum<<<gE1, BT, 0, stream>>>(tbuf, dstI, mmax, ssum, N_EDGES);
  gat_edge_aggregate<<<gE16, BT, 0, stream>>>(tbuf, srcI, dstI, mmax, ssum, hA, acc, N_EDGES);
  gat_bias_copy<<<gInit, BT, 0, stream>>>(acc, b0, hprev, NH);

  // ---------------- Layer 1 ----------------
  gat_prep<<<1, 64, 0, stream>>>(W1, as1, ad1, We1, ae1, wsv, wdv, wvv);
  gat_init<<<gInit, BT, 0, stream>>>(acc, mmax, ssum, N_NODES);
  gat_gemm_node<<<gGemm, BT, 0, stream>>>(hprev, W1, wsv, wdv, hA, a_s, a_d, N_NODES);
  gat_edge_logits<<<gE8, BT, 0, stream>>>(eattr, wvv, srcI, dstI, a_s, a_d, tbuf, mmax, N_EDGES);
  gat_edge_expsum<<<gE1, BT, 0, stream>>>(tbuf, dstI, mmax, ssum, N_EDGES);
  gat_edge_aggregate<<<gE16, BT, 0, stream>>>(tbuf, srcI, dstI, mmax, ssum, hA, acc, N_EDGES);

  // ---------------- Residual + mean pool ----------------
  pool_init<<<gPoolInit, BT, 0, stream>>>(pooled, counts);
  gat_pool<<<gN16, BT, 0, stream>>>(hprev, acc, b1, batch, pooled, counts, N_NODES);
  pool_final<<<gFinal, BT, 0, stream>>>(pooled, counts, out);
}